// KTVLoss_49486613184721
// MI455X (gfx1250) — compile-verified
//
#include <hip/hip_runtime.h>

typedef __attribute__((ext_vector_type(2))) float v2f;
typedef __attribute__((ext_vector_type(8))) float v8f;

#define IMG_N   24          // 8*3 images
#define HW      512
#define KW      10
#define XR      502         // box(|d/dH|) rows
#define XC      503         // box(|d/dH|) cols
#define YR      503         // box(|d/dW|) rows
#define YC      502         // box(|d/dW|) cols
#define STRIP   16
#define NSTRIP  32          // ceil(502/16)
#define NBLK    (IMG_N*NSTRIP)
#define TPB     512

// ---- dynamic LDS layout (float offsets) ----
#define HB_ROWS 28
#define OFF_HBS 0
#define OFF_HBT (OFF_HBS + HB_ROWS*512)
#define OFF_XS  (OFF_HBT + HB_ROWS*512)
#define OFF_XT  (OFF_XS  + 16*512)
#define OFF_YS  (OFF_XT  + 16*512)
#define OFF_YT  (OFF_YS  + 17*512)
#define OFF_SR  (OFF_YT  + 17*512)
#define OFF_ST  (OFF_SR  + 512)
#define OFF_RB  (OFF_ST  + 512)
#define SMEM_FLOATS (OFF_RB + 3*512)
#define SMEM_BYTES  (SMEM_FLOATS * 4)   // 260096 B < 320KB/WGP

__global__ void __launch_bounds__(TPB)
ktv_main(const float* __restrict__ L, const float* __restrict__ R,
         const float* __restrict__ I, float* __restrict__ part) {
  extern __shared__ float sm[];
  float* hbS = sm + OFF_HBS;   // horizontal box sums, S = |g_l|+|g_r|
  float* hbT = sm + OFF_HBT;   // horizontal box sums, T = |g_i|
  float* XS  = sm + OFF_XS;    // staged X field (16 rows)
  float* XT  = sm + OFF_XT;
  float* YS  = sm + OFF_YS;    // staged Y field (17 rows)
  float* YT  = sm + OFF_YT;
  float* srS = sm + OFF_SR;    // row scratch
  float* srT = sm + OFF_ST;
  float* rb  = sm + OFF_RB;    // 3x512 row buffer (pass B)

  const int t     = threadIdx.x;
  const int img   = blockIdx.x / NSTRIP;
  const int strip = blockIdx.x % NSTRIP;
  const int r0    = strip * STRIP;
  const int nX    = min(STRIP, XR - r0);                 // valid X rows this strip
  const int yend  = ((r0 + nX) * XC - 1) / YC;           // last Y row touched
  const int nY    = yend - r0 + 1;                       // <= 17
  const size_t base = (size_t)img * (HW * HW);
  const float* Lp = L + base;
  const float* Rp = R + base;
  const float* Ip = I + base;

  // zero hbuf: rows beyond the written range must be finite (zero) for WMMA
  for (int i = t; i < 2 * HB_ROWS * 512; i += TPB) hbS[i] = 0.0f;

  // ---- per-lane banded A-matrix weights for V_WMMA_F32_16X16X4_F32 ----
  // A is 16x4 (MxK): lane holds M = lane%16; VGPR0 K = 2*(lane/16), VGPR1 K+1.
  // out[m] = sum_{k in [m, m+9]} in[k]  ->  A[m][k] = (m <= k <= m+9)
  const int lane  = t & 31;
  const int mrow  = lane & 15;
  const int khalf = (lane >> 4) << 1;
  const int hi    = lane >> 4;
  float w0[7], w1[7];
#pragma unroll
  for (int q = 0; q < 7; ++q) {
    int k0 = 4 * q + khalf;
    int k1 = k0 + 1;
    w0[q] = (k0 >= mrow && k0 <= mrow + 9) ? 1.0f : 0.0f;
    w1[q] = (k1 >= mrow && k1 <= mrow + 9) ? 1.0f : 0.0f;
  }

  float gxAcc = 0.0f, gyAcc = 0.0f;
  __syncthreads();

  // ================= Pass A: gx = d/dH, horizontal 1x10 box sums =================
  float pl = Lp[(size_t)r0 * HW + t];
  float pr = Rp[(size_t)r0 * HW + t];
  float pi = Ip[(size_t)r0 * HW + t];
  const int rowsA = nX + 10;                  // img rows r0 .. r0+nX+9 (<= 511)
  for (int k = 1; k < rowsA; ++k) {
    const int h = r0 + k;
    const size_t ro = (size_t)h * HW + t;
    if (k + 1 < rowsA) {                      // prefetch next streamed row
      __builtin_prefetch(&Lp[ro + HW]);
      __builtin_prefetch(&Rp[ro + HW]);
      __builtin_prefetch(&Ip[ro + HW]);
    }
    float cl = Lp[ro], cr = Rp[ro], ci = Ip[ro];
    float dl = cl - pl, dr = cr - pr, di = ci - pi;
    const int g = h - 1;                      // gx row index
    if (g < r0 + STRIP && g < HW - 1) gxAcc += fabsf(dl + dr - di);
    srS[t] = fabsf(dl) + fabsf(dr);
    srT[t] = fabsf(di);
    __syncthreads();
    if (t < XC) {
      float sS = 0.0f, sT = 0.0f;
#pragma unroll
      for (int d = 0; d < KW; ++d) { sS += srS[t + d]; sT += srT[t + d]; }
      hbS[(k - 1) * 512 + t] = sS;
      hbT[(k - 1) * 512 + t] = sT;
    }
    __syncthreads();
    pl = cl; pr = cr; pi = ci;
  }

  // ---- WMMA vertical 10-row sums: 16 waves x 4 jobs = 32 col-tiles x {S,T} ----
  {
    const int wave = t >> 5;
#pragma unroll
    for (int j = 0; j < 4; ++j) {
      const int job = wave * 4 + j;                 // 0..63
      const float* hb = (job >> 5) ? hbT : hbS;
      float* dst      = (job >> 5) ? XT  : XS;
      const int ctile = (job & 31) * 16;
      const int col   = ctile + mrow;               // B: lane holds N = lane%16
      v8f acc = {0.f, 0.f, 0.f, 0.f, 0.f, 0.f, 0.f, 0.f};
      for (int q = 0; q < 7; ++q) {                 // 25 input rows in 7 K=4 chunks
        const int rr = 4 * q + khalf;
        v2f a; a.x = w0[q]; a.y = w1[q];
        v2f b; b.x = hb[rr * 512 + col]; b.y = hb[(rr + 1) * 512 + col];
        acc = __builtin_amdgcn_wmma_f32_16x16x4_f32(false, a, false, b,
                                                    (short)0, acc, false, false);
      }
#pragma unroll
      for (int v = 0; v < 8; ++v)                   // C/D layout: M = v + 8*hi
        dst[(v + 8 * hi) * 512 + ctile + mrow] = acc[v];
    }
  }
  __syncthreads();

  // ================= Pass B: gy = d/dW, horizontal 1x10 box sums =================
  const int rowsB = nY + 9;                   // img rows r0 .. r0+nY+8 (<= 511)
  for (int k = 0; k < rowsB; ++k) {
    const int h = r0 + k;
    const size_t ro = (size_t)h * HW + t;
    if (k + 1 < rowsB) {
      __builtin_prefetch(&Lp[ro + HW]);
      __builtin_prefetch(&Rp[ro + HW]);
      __builtin_prefetch(&Ip[ro + HW]);
    }
    rb[t] = Lp[ro]; rb[512 + t] = Rp[ro]; rb[1024 + t] = Ip[ro];
    __syncthreads();
    if (t < HW - 1) {
      float gl = rb[t + 1] - rb[t];
      float gr = rb[512 + t + 1] - rb[512 + t];
      float gi = rb[1024 + t + 1] - rb[1024 + t];
      if (h < r0 + STRIP) gyAcc += fabsf(gl + gr - gi);
      srS[t] = fabsf(gl) + fabsf(gr);
      srT[t] = fabsf(gi);
    }
    __syncthreads();
    if (t < YC) {
      float sS = 0.0f, sT = 0.0f;
#pragma unroll
      for (int d = 0; d < KW; ++d) { sS += srS[t + d]; sT += srT[t + d]; }
      hbS[k * 512 + t] = sS;
      hbT[k * 512 + t] = sT;
    }
    __syncthreads();
  }

  // ---- WMMA vertical sums -> Y stage (second row-tile only when nY == 17) ----
  {
    const int wave = t >> 5;
    const int nrt = (nY > 16) ? 2 : 1;
    for (int rt = 0; rt < nrt; ++rt) {
      const int rbase = rt * 16;
      const int nch   = (rt == 0) ? 7 : 3;          // rt=1: rows 16..27 only
#pragma unroll
      for (int j = 0; j < 4; ++j) {
        const int job = wave * 4 + j;
        const float* hb = (job >> 5) ? hbT : hbS;
        float* dst      = (job >> 5) ? YT  : YS;
        const int ctile = (job & 31) * 16;
        const int col   = ctile + mrow;
        v8f acc = {0.f, 0.f, 0.f, 0.f, 0.f, 0.f, 0.f, 0.f};
        for (int q = 0; q < nch; ++q) {
          const int rr = rbase + 4 * q + khalf;
          v2f a; a.x = w0[q]; a.y = w1[q];
          v2f b; b.x = hb[rr * 512 + col]; b.y = hb[(rr + 1) * 512 + col];
          acc = __builtin_amdgcn_wmma_f32_16x16x4_f32(false, a, false, b,
                                                      (short)0, acc, false, false);
        }
#pragma unroll
        for (int v = 0; v < 8; ++v) {
          const int row = rbase + v + 8 * hi;
          if (row < 17) dst[row * 512 + ctile + mrow] = acc[v];
        }
      }
    }
  }
  __syncthreads();

  // ================= Pass C: flat-index ratio sum =================
  float rAcc = 0.0f;
  {
    const int count = nX * XC;
    const int fbase = r0 * XC;
    for (int j = t; j < count; j += TPB) {
      const int rX = j / XC;
      const int cX = j - rX * XC;
      const int i  = fbase + j;
      const int rY = i / YC;
      const int cY = i - rY * YC;
      const int rYr = rY - r0;                     // in [0, nY)
      const float num = XS[rX * 512 + cX] + YS[rYr * 512 + cY];
      const float den = XT[rX * 512 + cX] + YT[rYr * 512 + cY] + 1e-4f; // 100*eps
      rAcc += num / den;
    }
  }

  // ---- deterministic block reduction (no atomics) ----
  float* red = srS;                                // srS|srT|rb contiguous: 3*512
  __syncthreads();
  red[t] = rAcc; red[512 + t] = gxAcc; red[1024 + t] = gyAcc;
  __syncthreads();
  for (int s = 256; s > 0; s >>= 1) {
    if (t < s) {
      red[t] += red[t + s];
      red[512 + t] += red[512 + t + s];
      red[1024 + t] += red[1024 + t + s];
    }
    __syncthreads();
  }
  if (t == 0) {
    part[blockIdx.x * 3 + 0] = red[0];
    part[blockIdx.x * 3 + 1] = red[512];
    part[blockIdx.x * 3 + 2] = red[1024];
  }
}

__global__ void __launch_bounds__(256)
ktv_finalize(const float* __restrict__ part, float* __restrict__ out) {
  __shared__ float smf[3 * 256];
  const int t = threadIdx.x;
  float a = 0.f, b = 0.f, c = 0.f;
  for (int i = t; i < NBLK; i += 256) {
    a += part[3 * i + 0];
    b += part[3 * i + 1];
    c += part[3 * i + 2];
  }
  smf[t] = a; smf[256 + t] = b; smf[512 + t] = c;
  __syncthreads();
  for (int s = 128; s > 0; s >>= 1) {
    if (t < s) {
      smf[t] += smf[t + s];
      smf[256 + t] += smf[256 + t + s];
      smf[512 + t] += smf[512 + t + s];
    }
    __syncthreads();
  }
  if (t == 0) {
    const float norm_loss = smf[0] / (float)(IMG_N * (XR * XC));     // /6,060,144
    const float grad_loss = smf[256] / (float)(IMG_N * 511 * 512) +
                            smf[512] / (float)(IMG_N * 512 * 511);
    out[0] = 1e-4f * norm_loss + grad_loss;
  }
}

extern "C" void kernel_launch(void* const* d_in, const int* in_sizes, int n_in,
                              void* d_out, int out_size, void* d_ws, size_t ws_size,
                              hipStream_t stream) {
  (void)in_sizes; (void)n_in; (void)out_size; (void)ws_size;
  const float* out_l   = (const float*)d_in[0];
  const float* out_r   = (const float*)d_in[1];
  const float* input_i = (const float*)d_in[2];
  float* part = (float*)d_ws;                      // 768*3 floats of partials

  // allow ~254KB of dynamic LDS (320KB/WGP on CDNA5)
  (void)hipFuncSetAttribute(reinterpret_cast<const void*>(ktv_main),
                            hipFuncAttributeMaxDynamicSharedMemorySize, SMEM_BYTES);

  ktv_main<<<NBLK, TPB, SMEM_BYTES, stream>>>(out_l, out_r, input_i, part);
  ktv_finalize<<<1, 256, 0, stream>>>(part, (float*)d_out);
}